// ESN_62912680952218
// MI455X (gfx1250) — compile-verified
//
#include <hip/hip_runtime.h>
#include <math.h>

// ESN forward: 512 serial steps of H = 0.5*H + 0.5*tanh(x_t@W_in + H@W_res),
// then readout H@W_out + b_out.
//  - f32 WMMA (v_wmma_f32_16x16x4_f32): full reference precision; compute-bound
//    per-step GEMM with W_res (16MB) L2-resident (192MB L2), HBM traffic ~50MB total.
//  - Double-buffered LDS filled by GLOBAL_LOAD_ASYNC_TO_LDS_B128 (ASYNCcnt),
//    overlapping the next K-tile fetch with the current tile's 32 WMMAs.
//
// Workspace: [0 .. 256*2048) = h buffer A, [256*2048 .. 2*256*2048) = h buffer B.
// Requires ws_size >= 4 MiB.

typedef __attribute__((ext_vector_type(2))) float v2f;
typedef __attribute__((ext_vector_type(8))) float v8f;

#define ESN_BATCH 256
#define ESN_SEQ   512
#define ESN_NIN   64
#define ESN_RES   2048
#define ESN_NOUT  10

#define BM 64
#define BN 64
#define BK 32
#define LDT 36   // A tile LDS row stride (floats): conflict-free frag reads, 16B-aligned b128
#define BNP 72   // B tile LDS row stride (floats): lane-halves hit disjoint bank ranges

__global__ __launch_bounds__(128) void esn_step_kernel(
    const float* __restrict__ x_t,   // (BATCH, NIN) rows strided by strideX floats
    int strideX,
    const float* __restrict__ W_in,  // (NIN, RES) row-major
    const float* __restrict__ W_res, // (RES, RES) row-major
    const float* __restrict__ h_in,  // (BATCH, RES)
    float* __restrict__ h_out)       // (BATCH, RES)
{
  __shared__ float lA[2][BM * LDT];  // 2 x 9216 B
  __shared__ float lB[2][BK * BNP];  // 2 x 9216 B

  const int tid  = threadIdx.x;
  const int lane = tid & 31;
  const int wv   = tid >> 5;            // wave id 0..3
  const int row0 = blockIdx.y * BM;     // batch-tile base
  const int col0 = blockIdx.x * BN;     // reservoir-tile base
  const int wm   = (wv >> 1) * 32;      // wave row offset in tile
  const int wn   = (wv & 1) * 32;       // wave col offset in tile
  const int m16  = lane & 15;
  const int kp   = (lane >> 4) * 2;     // K pair base per ISA A/B fragment layout

  v8f acc[2][2];
#pragma unroll
  for (int i = 0; i < 2; ++i)
#pragma unroll
    for (int j = 0; j < 2; ++j)
      acc[i][j] = (v8f){0.f, 0.f, 0.f, 0.f, 0.f, 0.f, 0.f, 0.f};

  // Flattened K pipeline: blocks 0..1 = x_t@W_in (K=64), blocks 2..65 = h@W_res (K=2048)
  const int NKB = 2 + ESN_RES / BK;  // 66

  // Issue async global->LDS copies for K-block `blk` into LDS buffer `bsel`.
  auto issue_tile = [&](int blk, int bsel) {
    const float* Ag;
    const float* Bg;
    int As, kb;
    if (blk < 2) { Ag = x_t;  As = strideX;  Bg = W_in;  kb = blk * BK; }
    else         { Ag = h_in; As = ESN_RES;  Bg = W_res; kb = (blk - 2) * BK; }
    // A tile: BM x BK row-major (padded LDT), 4 x b128 per thread
#pragma unroll
    for (int i = 0; i < 4; ++i) {
      int idx = tid + i * 128;                 // 0..511
      int r   = idx >> 3;                      // 0..63
      int c4  = (idx & 7) << 2;                // 0..28
      unsigned dst = (unsigned)(unsigned long long)&lA[bsel][r * LDT + c4];
      unsigned long long src = (unsigned long long)(Ag + (size_t)(row0 + r) * As + kb + c4);
      asm volatile("global_load_async_to_lds_b128 %0, %1, off"
                   :: "v"(dst), "v"(src) : "memory");
    }
    // B tile: BK x BN row-major (padded BNP), 4 x b128 per thread
#pragma unroll
    for (int i = 0; i < 4; ++i) {
      int idx = tid + i * 128;                 // 0..511
      int kk  = idx >> 4;                      // 0..31
      int n4  = (idx & 15) << 2;               // 0..60
      unsigned dst = (unsigned)(unsigned long long)&lB[bsel][kk * BNP + n4];
      unsigned long long src = (unsigned long long)(Bg + (size_t)(kb + kk) * ESN_RES + col0 + n4);
      asm volatile("global_load_async_to_lds_b128 %0, %1, off"
                   :: "v"(dst), "v"(src) : "memory");
    }
  };

  issue_tile(0, 0);

#pragma unroll 1
  for (int b = 0; b < NKB; ++b) {
    const int cur = b & 1;
    asm volatile("s_wait_asynccnt 0x0" ::: "memory");
    __syncthreads();
    if (b + 1 < NKB) issue_tile(b + 1, cur ^ 1);

    const float* __restrict__ A = lA[cur];
    const float* __restrict__ B = lB[cur];
#pragma unroll
    for (int kk = 0; kk < BK; kk += 4) {
      const int k = kk + kp;
      v2f a0 = *(const v2f*)(&A[(wm + m16) * LDT + k]);
      v2f a1 = *(const v2f*)(&A[(wm + 16 + m16) * LDT + k]);
      v2f b0 = {B[(k + 0) * BNP + wn + m16],
                B[(k + 1) * BNP + wn + m16]};
      v2f b1 = {B[(k + 0) * BNP + wn + 16 + m16],
                B[(k + 1) * BNP + wn + 16 + m16]};
      acc[0][0] = __builtin_amdgcn_wmma_f32_16x16x4_f32(false, a0, false, b0, (short)0, acc[0][0], false, false);
      acc[0][1] = __builtin_amdgcn_wmma_f32_16x16x4_f32(false, a0, false, b1, (short)0, acc[0][1], false, false);
      acc[1][0] = __builtin_amdgcn_wmma_f32_16x16x4_f32(false, a1, false, b0, (short)0, acc[1][0], false, false);
      acc[1][1] = __builtin_amdgcn_wmma_f32_16x16x4_f32(false, a1, false, b1, (short)0, acc[1][1], false, false);
    }
  }

  // ---- fused leaky-tanh epilogue: h_out = 0.5*h_in + 0.5*tanh(acc) ----
  const int hhalf = (lane >> 4) * 8;
#pragma unroll
  for (int i = 0; i < 2; ++i) {
#pragma unroll
    for (int j = 0; j < 2; ++j) {
      union { v8f v; float f[8]; } u;
      u.v = acc[i][j];
      const int col   = col0 + wn + j * 16 + m16;
      const int rbase = row0 + wm + i * 16 + hhalf;
#pragma unroll
      for (int r = 0; r < 8; ++r) {
        const size_t off = (size_t)(rbase + r) * ESN_RES + col;
        h_out[off] = 0.5f * h_in[off] + 0.5f * tanhf(u.f[r]);
      }
    }
  }
}

// Readout: one wave per batch row. out[b,o] = sum_r h[b,r]*W_out[r,o] + b_out[o]
__global__ __launch_bounds__(32) void esn_readout_kernel(
    const float* __restrict__ h,      // (BATCH, RES)
    const float* __restrict__ W_out,  // (RES, NOUT)
    const float* __restrict__ b_out,  // (NOUT)
    float* __restrict__ out)          // (BATCH, NOUT)
{
  const int b    = blockIdx.x;
  const int lane = threadIdx.x;
  float sums[ESN_NOUT];
#pragma unroll
  for (int o = 0; o < ESN_NOUT; ++o) sums[o] = 0.f;

  for (int r = lane; r < ESN_RES; r += 32) {
    const float hv = h[(size_t)b * ESN_RES + r];
#pragma unroll
    for (int o = 0; o < ESN_NOUT; ++o)
      sums[o] += hv * W_out[(size_t)r * ESN_NOUT + o];
  }
#pragma unroll
  for (int off = 16; off > 0; off >>= 1) {
#pragma unroll
    for (int o = 0; o < ESN_NOUT; ++o)
      sums[o] += __shfl_down(sums[o], off, 32);
  }
  if (lane == 0) {
#pragma unroll
    for (int o = 0; o < ESN_NOUT; ++o)
      out[(size_t)b * ESN_NOUT + o] = sums[o] + b_out[o];
  }
}

extern "C" void kernel_launch(void* const* d_in, const int* in_sizes, int n_in,
                              void* d_out, int out_size, void* d_ws, size_t ws_size,
                              hipStream_t stream) {
  const float* x     = (const float*)d_in[0];  // (256, 512, 64)
  const float* W_in  = (const float*)d_in[1];  // (64, 2048)
  const float* W_res = (const float*)d_in[2];  // (2048, 2048)
  const float* W_out = (const float*)d_in[3];  // (2048, 10)
  const float* b_out = (const float*)d_in[4];  // (10,)
  float* out = (float*)d_out;

  float* h0 = (float*)d_ws;
  float* h1 = h0 + (size_t)ESN_BATCH * ESN_RES;

  // h starts at zero
  hipMemsetAsync(h0, 0, (size_t)ESN_BATCH * ESN_RES * sizeof(float), stream);

  dim3 grid(ESN_RES / BN, ESN_BATCH / BM);   // (32, 4)
  const int strideX = ESN_SEQ * ESN_NIN;     // row stride of x_t within x

  for (int t = 0; t < ESN_SEQ; ++t) {
    const float* hin  = (t & 1) ? h1 : h0;
    float*       hout = (t & 1) ? h0 : h1;
    esn_step_kernel<<<grid, 128, 0, stream>>>(
        x + (size_t)t * ESN_NIN, strideX, W_in, W_res, hin, hout);
  }
  // 512 steps: last write (t=511, odd) lands in h0
  esn_readout_kernel<<<ESN_BATCH, 32, 0, stream>>>(h0, W_out, b_out, out);
}